// TimeVaryingBiquad_10101763080719
// MI455X (gfx1250) — compile-verified
//
#include <hip/hip_runtime.h>

// ---------------- problem geometry (fixed by reference) ----------------
constexpr int B_ROWS      = 32;
constexpr int T_LEN       = 65536;
constexpr int CHUNK       = 16;                  // samples per thread (serial)
constexpr int NCH_ROW     = T_LEN / CHUNK;       // 4096 chunks per batch row
constexpr int CH_PER_BLK  = 256;                 // threads per block in K1/K3
constexpr int TILE        = CH_PER_BLK * CHUNK;  // 4096 samples per block tile
constexpr int BLKS_PER_ROW= NCH_ROW / CH_PER_BLK;// 16
constexpr int NCH_TOTAL   = B_ROWS * NCH_ROW;    // 131072
constexpr int SCAN_T      = 1024;                // threads in scan block
constexpr int CPT         = NCH_ROW / SCAN_T;    // 4 chunks per scan thread

// log-domain modulation constants (match reference)
#define LOG_MIN_W (-6.907755279f)   // ln(1e-3)
#define LOG_MAX_W ( 1.144411525f)   // ln(pi - 1e-3)
#define LOG_MIN_Q (-0.346583190f)   // ln(0.7071)
#define LOG_MAX_Q ( 1.386294361f)   // ln(4)

// ---------------- CDNA5 async global->LDS staging path ----------------
#if defined(__HIP_DEVICE_COMPILE__)
  #if __has_builtin(__builtin_amdgcn_global_load_async_to_lds_b128) && \
      __has_builtin(__builtin_amdgcn_s_wait_asynccnt)
    #define ASYNC_MODE 1   // clang builtins
  #else
    #define ASYNC_MODE 2   // inline gfx1250 asm
  #endif
#else
  #define ASYNC_MODE 0     // host pass: plain code only
#endif

#if ASYNC_MODE
typedef int v4i __attribute__((ext_vector_type(4)));
typedef __attribute__((address_space(1))) v4i gv4i;   // global int4
typedef __attribute__((address_space(3))) v4i lv4i;   // LDS int4
typedef __attribute__((address_space(3))) void ldsv_t;
#endif

__device__ __forceinline__ float rcp_f(float v) {
#if defined(__HIP_DEVICE_COMPILE__) && __has_builtin(__builtin_amdgcn_rcpf)
  return __builtin_amdgcn_rcpf(v);   // v_rcp_f32
#else
  return 1.0f / v;
#endif
}

// Stage one TILE (4096 f32 = 16 KiB) into LDS, fully coalesced (512 B / wave op).
__device__ __forceinline__ void stage_tile_async(const float* gsrc, float* ldst, int tid) {
#if ASYNC_MODE == 1
  #pragma unroll
  for (int k = 0; k < 4; ++k) {
    const int off = (k * CH_PER_BLK + tid) * 16;  // bytes
    char* g = (char*)gsrc + off;                  // drop const (builtin is non-const)
    char* l = (char*)ldst + off;
    __builtin_amdgcn_global_load_async_to_lds_b128((gv4i*)g, (lv4i*)l, 0, 0);
  }
#elif ASYNC_MODE == 2
  #pragma unroll
  for (int k = 0; k < 4; ++k) {
    const int off = (k * CH_PER_BLK + tid) * 16;
    unsigned lds_off = (unsigned)(unsigned long long)(ldsv_t*)((char*)ldst + off);
    unsigned long long gaddr = (unsigned long long)((const char*)gsrc + off);
    asm volatile("global_load_async_to_lds_b128 %0, %1, off"
                 :: "v"(lds_off), "v"(gaddr) : "memory");
  }
#else
  (void)gsrc; (void)ldst; (void)tid;
#endif
}

__device__ __forceinline__ void wait_async_and_sync() {
#if ASYNC_MODE == 1
  __builtin_amdgcn_s_wait_asynccnt(0);
  __syncthreads();
#elif ASYNC_MODE == 2
  asm volatile("s_wait_asynccnt 0" ::: "memory");
  __syncthreads();
#endif
}

// ---------------- per-sample coefficient math ----------------
__device__ __forceinline__ void biquad_coeffs(float mw, float mq,
                                              float& a1, float& a2,
                                              float& b0, float& b1) {
  float w  = __expf(LOG_MIN_W + (LOG_MAX_W - LOG_MIN_W) * mw);
  float q  = __expf(LOG_MIN_Q + (LOG_MAX_Q - LOG_MIN_Q) * mq);
  float sw = __sinf(w);
  float cw = __cosf(w);
  float alpha  = sw * 0.5f * rcp_f(q);
  float inv_a0 = rcp_f(1.0f + alpha);
  a1 = -2.0f * cw * inv_a0;
  a2 = (1.0f - alpha) * inv_a0;
  b1 = (1.0f - cw) * inv_a0;
  b0 = 0.5f * b1;
}

// ---------------- K1: per-chunk affine summaries ----------------
// state s_t = (y_t, y_{t-1}); chunk maps s_in -> A*s_in + b
__global__ __launch_bounds__(CH_PER_BLK) void tvb_k1_summaries(
    const float* __restrict__ x, const float* __restrict__ cut,
    const float* __restrict__ res, float* __restrict__ ws)
{
  const int tid  = threadIdx.x;
  const int row  = blockIdx.x / BLKS_PER_ROW;
  const int tile = blockIdx.x % BLKS_PER_ROW;
  const int base = row * T_LEN + tile * TILE;

#if ASYNC_MODE
  __shared__ float s_x[TILE], s_c[TILE], s_r[TILE];
  stage_tile_async(x + base,   s_x, tid);
  stage_tile_async(cut + base, s_c, tid);
  stage_tile_async(res + base, s_r, tid);
  wait_async_and_sync();
  const float4* xv = (const float4*)s_x;
  const float4* cv = (const float4*)s_c;
  const float4* rv = (const float4*)s_r;
#else
  const float4* xv = (const float4*)(x + base);
  const float4* cv = (const float4*)(cut + base);
  const float4* rv = (const float4*)(res + base);
#endif
  const int voff = tid * (CHUNK / 4);

  float A00 = 1.f, A01 = 0.f, A10 = 0.f, A11 = 0.f;
  float sb0 = 0.f, sb1 = 0.f;
  #pragma unroll
  for (int k = 0; k < CHUNK / 4; ++k) {
    const float4 xq = xv[voff + k], cq = cv[voff + k], rq = rv[voff + k];
    const float xs[4] = {xq.x, xq.y, xq.z, xq.w};
    const float cs[4] = {cq.x, cq.y, cq.z, cq.w};
    const float rs[4] = {rq.x, rq.y, rq.z, rq.w};
    #pragma unroll
    for (int e = 0; e < 4; ++e) {
      float a1, a2, b0, b1;
      biquad_coeffs(cs[e], rs[e], a1, a2, b0, b1);
      // compose companion matrix M=[[-a1,-a2],[1,0]] on the left (cheap: row shift)
      float n00 = -a1 * A00 - a2 * A10;
      float n01 = -a1 * A01 - a2 * A11;
      A10 = A00; A11 = A01; A00 = n00; A01 = n01;
      float nb0 = -a1 * sb0 - a2 * sb1 + xs[e];
      sb1 = sb0; sb0 = nb0;
    }
  }
  const int cid = row * NCH_ROW + tile * CH_PER_BLK + tid;
  ws[0 * NCH_TOTAL + cid] = A00;
  ws[1 * NCH_TOTAL + cid] = A01;
  ws[2 * NCH_TOTAL + cid] = A10;
  ws[3 * NCH_TOTAL + cid] = A11;
  ws[4 * NCH_TOTAL + cid] = sb0;
  ws[5 * NCH_TOTAL + cid] = sb1;
}

// ---------------- K2: per-row scan of affine summaries ----------------
struct Aff { float a00, a01, a10, a11, b0, b1; };

__device__ __forceinline__ Aff aff_comb(const Aff& P, const Aff& Q) { // Q after P
  Aff r;
  r.a00 = Q.a00 * P.a00 + Q.a01 * P.a10;
  r.a01 = Q.a00 * P.a01 + Q.a01 * P.a11;
  r.a10 = Q.a10 * P.a00 + Q.a11 * P.a10;
  r.a11 = Q.a10 * P.a01 + Q.a11 * P.a11;
  r.b0  = Q.a00 * P.b0  + Q.a01 * P.b1 + Q.b0;
  r.b1  = Q.a10 * P.b0  + Q.a11 * P.b1 + Q.b1;
  return r;
}

__global__ __launch_bounds__(SCAN_T) void tvb_k2_scan(float* __restrict__ ws)
{
  __shared__ Aff s[SCAN_T];   // 24 KiB
  const int row   = blockIdx.x;
  const int tid   = threadIdx.x;
  const int cbase = row * NCH_ROW + tid * CPT;

  Aff S[CPT];
  #pragma unroll
  for (int j = 0; j < CPT; ++j) {
    const int c = cbase + j;
    S[j].a00 = ws[0*NCH_TOTAL + c]; S[j].a01 = ws[1*NCH_TOTAL + c];
    S[j].a10 = ws[2*NCH_TOTAL + c]; S[j].a11 = ws[3*NCH_TOTAL + c];
    S[j].b0  = ws[4*NCH_TOTAL + c]; S[j].b1  = ws[5*NCH_TOTAL + c];
  }
  Aff cur = S[0];
  #pragma unroll
  for (int j = 1; j < CPT; ++j) cur = aff_comb(cur, S[j]);

  s[tid] = cur;
  __syncthreads();
  for (int off = 1; off < SCAN_T; off <<= 1) {   // Hillis–Steele inclusive scan
    Aff other;
    const bool vld = (tid >= off);
    if (vld) other = s[tid - off];
    __syncthreads();
    if (vld) { cur = aff_comb(other, cur); s[tid] = cur; }
    __syncthreads();
  }

  // exclusive prefix applied to the zero initial state -> start state = prefix.b
  float y1 = 0.f, y2 = 0.f;
  if (tid > 0) { Aff p = s[tid - 1]; y1 = p.b0; y2 = p.b1; }
  #pragma unroll
  for (int j = 0; j < CPT; ++j) {
    const int c = cbase + j;
    ws[6*NCH_TOTAL + c] = y1;   // y_a[t0-1]
    ws[7*NCH_TOTAL + c] = y2;   // y_a[t0-2]
    const float ny1 = S[j].a00 * y1 + S[j].a01 * y2 + S[j].b0;
    const float ny2 = S[j].a10 * y1 + S[j].a11 * y2 + S[j].b1;
    y1 = ny1; y2 = ny2;
  }
}

// ---------------- K3: apply recurrence + fused FIR ----------------
__global__ __launch_bounds__(CH_PER_BLK) void tvb_k3_apply(
    const float* __restrict__ x, const float* __restrict__ cut,
    const float* __restrict__ res, const float* __restrict__ ws,
    float* __restrict__ out)
{
  const int tid  = threadIdx.x;
  const int row  = blockIdx.x / BLKS_PER_ROW;
  const int tile = blockIdx.x % BLKS_PER_ROW;
  const int base = row * T_LEN + tile * TILE;

#if ASYNC_MODE
  __shared__ float s_x[TILE], s_c[TILE], s_r[TILE];
  stage_tile_async(x + base,   s_x, tid);
  stage_tile_async(cut + base, s_c, tid);
  stage_tile_async(res + base, s_r, tid);
  wait_async_and_sync();
  const float4* xv = (const float4*)s_x;
  const float4* cv = (const float4*)s_c;
  const float4* rv = (const float4*)s_r;
#else
  const float4* xv = (const float4*)(x + base);
  const float4* cv = (const float4*)(cut + base);
  const float4* rv = (const float4*)(res + base);
#endif
  const int voff = tid * (CHUNK / 4);
  const int cid  = row * NCH_ROW + tile * CH_PER_BLK + tid;

  float y1 = ws[6*NCH_TOTAL + cid];   // y_a[t-1] entering the chunk
  float y2 = ws[7*NCH_TOTAL + cid];   // y_a[t-2]

  float4* ov = (float4*)(out + base);
  #pragma unroll
  for (int k = 0; k < CHUNK / 4; ++k) {
    const float4 xq = xv[voff + k], cq = cv[voff + k], rq = rv[voff + k];
    const float xs[4] = {xq.x, xq.y, xq.z, xq.w};
    const float cs[4] = {cq.x, cq.y, cq.z, cq.w};
    const float rs[4] = {rq.x, rq.y, rq.z, rq.w};
    float os[4];
    #pragma unroll
    for (int e = 0; e < 4; ++e) {
      float a1, a2, b0, b1;
      biquad_coeffs(cs[e], rs[e], a1, a2, b0, b1);
      const float ya = xs[e] - a1 * y1 - a2 * y2;            // all-pole IIR
      os[e] = b0 * (y2 + ya) + b1 * y1;                      // 3-tap FIR (b2==b0)
      y2 = y1; y1 = ya;
    }
    ov[voff + k] = make_float4(os[0], os[1], os[2], os[3]);
  }
}

// ---------------- launcher ----------------
extern "C" void kernel_launch(void* const* d_in, const int* in_sizes, int n_in,
                              void* d_out, int out_size, void* d_ws, size_t ws_size,
                              hipStream_t stream) {
  const float* x   = (const float*)d_in[0];
  const float* cut = (const float*)d_in[1];
  const float* res = (const float*)d_in[2];
  float* out = (float*)d_out;
  float* ws  = (float*)d_ws;   // needs 8 * 131072 * 4 B = 4 MiB

  const dim3 blkA(CH_PER_BLK);
  const dim3 grdA(B_ROWS * BLKS_PER_ROW);      // 512 blocks
  tvb_k1_summaries<<<grdA, blkA, 0, stream>>>(x, cut, res, ws);
  tvb_k2_scan<<<dim3(B_ROWS), dim3(SCAN_T), 0, stream>>>(ws);
  tvb_k3_apply<<<grdA, blkA, 0, stream>>>(x, cut, res, ws, out);
}